// Magnet_batch_29454885716513
// MI455X (gfx1250) — compile-verified
//
#include <hip/hip_runtime.h>
#include <hip/hip_bf16.h>

typedef float v2f __attribute__((ext_vector_type(2)));
typedef float v8f __attribute__((ext_vector_type(8)));

#ifndef PI_OVER_2F
#define PI_OVER_2F 1.5707963267948966f
#endif

// ---------------------------------------------------------------------------
// 1) init: deg = 0, t1r = t1i = 0, t2r = -xr, t2i = -xi
// ---------------------------------------------------------------------------
__global__ void magnet_init_kernel(const float* __restrict__ xr,
                                   const float* __restrict__ xi,
                                   float* __restrict__ deg,
                                   float* __restrict__ t1r, float* __restrict__ t1i,
                                   float* __restrict__ t2r, float* __restrict__ t2i,
                                   int n_nodes) {
    int idx = blockIdx.x * blockDim.x + threadIdx.x;
    int total = n_nodes * 32;
    if (idx < total) {
        t1r[idx] = 0.0f;
        t1i[idx] = 0.0f;
        t2r[idx] = -xr[idx];
        t2i[idx] = -xi[idx];
    }
    if (idx < n_nodes) deg[idx] = 0.0f;
}

// ---------------------------------------------------------------------------
// 2) degree: deg[src] += 0.5*w, deg[dst] += 0.5*w (w = src!=dst)
// ---------------------------------------------------------------------------
__global__ void magnet_degree_kernel(const int* __restrict__ src,
                                     const int* __restrict__ dst,
                                     float* __restrict__ deg, int n_edges) {
    int e = blockIdx.x * blockDim.x + threadIdx.x;
    if (e >= n_edges) return;
    int s = src[e], d = dst[e];
    if (s != d) {
        __hip_atomic_fetch_add(&deg[s], 0.5f, __ATOMIC_RELAXED, __HIP_MEMORY_SCOPE_AGENT);
        __hip_atomic_fetch_add(&deg[d], 0.5f, __ATOMIC_RELAXED, __HIP_MEMORY_SCOPE_AGENT);
    }
}

// ---------------------------------------------------------------------------
// 3) dis = deg > 0 ? rsqrt(max(deg,1e-30)) : 0   (in place)
// ---------------------------------------------------------------------------
__global__ void magnet_rsqrt_kernel(float* __restrict__ deg, int n_nodes) {
    int n = blockIdx.x * blockDim.x + threadIdx.x;
    if (n >= n_nodes) return;
    float dv = deg[n];
    deg[n] = (dv > 0.0f) ? __frsqrt_rn(fmaxf(dv, 1e-30f)) : 0.0f;
}

// ---------------------------------------------------------------------------
// 4) per-edge weights: norm = dis[s]*0.5*w*dis[d]; theta = (pi/2)*w
//    wr = -norm*cos(theta), wi = -norm*sin(theta)
//    (reverse direction: same wr, negated wi)
// ---------------------------------------------------------------------------
__global__ void magnet_edgew_kernel(const int* __restrict__ src,
                                    const int* __restrict__ dst,
                                    const float* __restrict__ dis,
                                    float* __restrict__ wr, float* __restrict__ wi,
                                    int n_edges) {
    int e = blockIdx.x * blockDim.x + threadIdx.x;
    if (e >= n_edges) return;
    int s = src[e], d = dst[e];
    float w = (s != d) ? 1.0f : 0.0f;
    float norm = dis[s] * (0.5f * w) * dis[d];
    float theta = PI_OVER_2F * w;
    wr[e] = -norm * cosf(theta);
    wi[e] = -norm * sinf(theta);
}

// ---------------------------------------------------------------------------
// 5) prop: one thread per (edge, feature). Wave = one edge x 32 features,
//    so gathers and atomics are coalesced 128B lines.
//    outR[d] += scale*wr*xr[s] ; outR[s] += scale*wr*xr[d]
//    outI[d] += scale*wi*xi[s] ; outI[s] -= scale*wi*xi[d]
// ---------------------------------------------------------------------------
__global__ void magnet_prop_kernel(const float* __restrict__ xr,
                                   const float* __restrict__ xi,
                                   float* __restrict__ outR, float* __restrict__ outI,
                                   const int* __restrict__ src, const int* __restrict__ dst,
                                   const float* __restrict__ wr, const float* __restrict__ wi,
                                   int n_edges, float scale) {
    unsigned int tid = blockIdx.x * blockDim.x + threadIdx.x;
    unsigned int total = (unsigned int)n_edges * 32u;
    if (tid >= total) return;
    int f = (int)(tid & 31u);
    int e = (int)(tid >> 5);
    int s = src[e], d = dst[e];
    float wrv = wr[e] * scale;
    float wiv = wi[e] * scale;
    size_t so = (size_t)s * 32 + f;
    size_t dof = (size_t)d * 32 + f;
    float ars = xr[so], ard = xr[dof];
    float ais = xi[so], aid = xi[dof];
    __hip_atomic_fetch_add(&outR[dof],  wrv * ars, __ATOMIC_RELAXED, __HIP_MEMORY_SCOPE_AGENT);
    __hip_atomic_fetch_add(&outR[so],   wrv * ard, __ATOMIC_RELAXED, __HIP_MEMORY_SCOPE_AGENT);
    __hip_atomic_fetch_add(&outI[dof],  wiv * ais, __ATOMIC_RELAXED, __HIP_MEMORY_SCOPE_AGENT);
    __hip_atomic_fetch_add(&outI[so],  -wiv * aid, __ATOMIC_RELAXED, __HIP_MEMORY_SCOPE_AGENT);
}

// ---------------------------------------------------------------------------
// 6) fused Chebyshev-mix GEMM via V_WMMA_F32_16X16X4_F32 (wave32).
//    One wave per 16-node tile:
//      rr = x@W0 + t1r@W1 + t2r@W2 ; ii = xi@W0 + t1i@W1 + t2i@W2
//      out_real = rr - ii + bias ; out_imag = rr + ii + bias
//    B lane layout per (term t, kstep kk, coltile nt):
//      lane<16 : {W[4kk+0][n], W[4kk+1][n]}, lane>=16 : {W[4kk+2][n], W[4kk+3][n]}
//    A lane layout: lane<16 : row cols {4kk,4kk+1}, lane>=16 : {4kk+2,4kk+3}
// ---------------------------------------------------------------------------
__global__ __launch_bounds__(256) void magnet_final_wmma_kernel(
    const float* __restrict__ xr,  const float* __restrict__ xi,
    const float* __restrict__ t1r, const float* __restrict__ t1i,
    const float* __restrict__ t2r, const float* __restrict__ t2i,
    const float* __restrict__ W,   // [3][32][32] row-major
    const float* __restrict__ bias,
    float* __restrict__ outR, float* __restrict__ outI,
    int n_nodes) {
    __shared__ float2 bpack[3 * 8 * 2 * 32];   // 12 KB, B operands pre-swizzled
    __shared__ float  sbias[32];

    int tid = threadIdx.x;
    for (int idx = tid; idx < 3 * 8 * 2 * 32; idx += 256) {
        int lane = idx & 31;
        int nt   = (idx >> 5) & 1;
        int kk   = (idx >> 6) & 7;
        int t    = idx >> 9;
        int k0   = kk * 4 + 2 * (lane >> 4);
        int n    = nt * 16 + (lane & 15);
        bpack[idx] = make_float2(W[t * 1024 + k0 * 32 + n],
                                 W[t * 1024 + (k0 + 1) * 32 + n]);
    }
    if (tid < 32) sbias[tid] = bias[tid];
    __syncthreads();

    int wave = tid >> 5;
    int lane = tid & 31;
    int tile = blockIdx.x * 8 + wave;
    int row0 = tile * 16;
    if (row0 >= n_nodes) return;          // whole wave exits together (EXEC stays all-1s)

    int half = lane >> 4;
    int r    = row0 + (lane & 15);
    int rl   = (r < n_nodes) ? r : (n_nodes - 1);   // clamp loads on ragged tail

    const float* Ar[3] = { xr, t1r, t2r };
    const float* Ai[3] = { xi, t1i, t2i };

    v8f accr0 = {}, accr1 = {}, acci0 = {}, acci1 = {};

    for (int t = 0; t < 3; ++t) {
        const float* pr = Ar[t] + (size_t)rl * 32 + half * 2;
        const float* pi = Ai[t] + (size_t)rl * 32 + half * 2;
#pragma unroll
        for (int kk = 0; kk < 8; ++kk) {
            v2f ar = *(const v2f*)(pr + kk * 4);
            v2f ai = *(const v2f*)(pi + kk * 4);
            v2f b0 = *(const v2f*)&bpack[((t * 8 + kk) * 2 + 0) * 32 + lane];
            v2f b1 = *(const v2f*)&bpack[((t * 8 + kk) * 2 + 1) * 32 + lane];
            accr0 = __builtin_amdgcn_wmma_f32_16x16x4_f32(false, ar, false, b0, (short)0, accr0, false, false);
            accr1 = __builtin_amdgcn_wmma_f32_16x16x4_f32(false, ar, false, b1, (short)0, accr1, false, false);
            acci0 = __builtin_amdgcn_wmma_f32_16x16x4_f32(false, ai, false, b0, (short)0, acci0, false, false);
            acci1 = __builtin_amdgcn_wmma_f32_16x16x4_f32(false, ai, false, b1, (short)0, acci1, false, false);
        }
    }

    // C/D layout: VGPR i -> row row0+i (lanes 0-15) / row0+i+8 (lanes 16-31)
    int orow = row0 + 8 * half;
    int c0   = lane & 15;
    int c1   = 16 + c0;
    float b0v = sbias[c0], b1v = sbias[c1];
#pragma unroll
    for (int i = 0; i < 8; ++i) {
        int rw = orow + i;
        if (rw >= n_nodes) continue;
        size_t o0 = (size_t)rw * 32 + c0;
        size_t o1 = (size_t)rw * 32 + c1;
        float rr0 = accr0[i], rr1 = accr1[i];
        float ii0 = acci0[i], ii1 = acci1[i];
        outR[o0] = rr0 - ii0 + b0v;
        outR[o1] = rr1 - ii1 + b1v;
        outI[o0] = rr0 + ii0 + b0v;
        outI[o1] = rr1 + ii1 + b1v;
    }
}

// ---------------------------------------------------------------------------
// launch
// ---------------------------------------------------------------------------
extern "C" void kernel_launch(void* const* d_in, const int* in_sizes, int n_in,
                              void* d_out, int out_size, void* d_ws, size_t ws_size,
                              hipStream_t stream) {
    const float* xr   = (const float*)d_in[0];
    const float* xi   = (const float*)d_in[1];
    const int*   eidx = (const int*)d_in[2];
    const float* W    = (const float*)d_in[3];
    const float* bias = (const float*)d_in[4];

    int n_nodes = in_sizes[0] / 32;
    int n_edges = in_sizes[2] / 2;
    const int* src = eidx;
    const int* dst = eidx + n_edges;

    // workspace carve-up (floats)
    float* ws  = (float*)d_ws;
    float* deg = ws;                                    // n_nodes (then reused as dis)
    float* wr  = deg + n_nodes;                         // n_edges
    float* wi  = wr + n_edges;                          // n_edges
    float* t1r = wi + n_edges;                          // n_nodes*32
    float* t1i = t1r + (size_t)n_nodes * 32;
    float* t2r = t1i + (size_t)n_nodes * 32;
    float* t2i = t2r + (size_t)n_nodes * 32;

    float* outR = (float*)d_out;                        // [n_nodes][32]
    float* outI = outR + (size_t)n_nodes * 32;

    const int B = 256;
    int gInit = (n_nodes * 32 + B - 1) / B;
    magnet_init_kernel<<<gInit, B, 0, stream>>>(xr, xi, deg, t1r, t1i, t2r, t2i, n_nodes);

    int gE = (n_edges + B - 1) / B;
    magnet_degree_kernel<<<gE, B, 0, stream>>>(src, dst, deg, n_edges);

    int gN = (n_nodes + B - 1) / B;
    magnet_rsqrt_kernel<<<gN, B, 0, stream>>>(deg, n_nodes);

    magnet_edgew_kernel<<<gE, B, 0, stream>>>(src, dst, deg, wr, wi, n_edges);

    unsigned int propThreads = (unsigned int)n_edges * 32u;
    unsigned int gProp = (propThreads + B - 1) / B;
    // pass 1: t1 = prop(x)
    magnet_prop_kernel<<<gProp, B, 0, stream>>>(xr, xi, t1r, t1i, src, dst, wr, wi, n_edges, 1.0f);
    // pass 2: t2 = 2*prop(t1) - x   (t2 pre-initialized to -x)
    magnet_prop_kernel<<<gProp, B, 0, stream>>>(t1r, t1i, t2r, t2i, src, dst, wr, wi, n_edges, 2.0f);

    int tiles  = (n_nodes + 15) / 16;
    int gFinal = (tiles + 7) / 8;       // 8 waves (tiles) per 256-thread block
    magnet_final_wmma_kernel<<<gFinal, B, 0, stream>>>(xr, xi, t1r, t1i, t2r, t2i,
                                                       W, bias, outR, outI, n_nodes);
}